// DCTTransform_40922448396795
// MI455X (gfx1250) — compile-verified
//
#include <hip/hip_runtime.h>
#include <hip/hip_bf16.h>
#include <math.h>

typedef __attribute__((ext_vector_type(2))) float v2f;
typedef __attribute__((ext_vector_type(8))) float v8f;

#define PATCH 32
#define BATCH 32
#define NPAT 256
#define CH 3
#define KFILT 4
#define TASKS (BATCH * NPAT)          // 8192 (b,n) wave-tasks
#define WAVES_PER_BLOCK 8
#define THREADS (WAVES_PER_BLOCK * 32)

// One wave computes the full 2D DCT (D @ X @ D^T) of one 32x32 patch per channel
// using V_WMMA_F32_16X16X4_F32 tiles, writes coefficients, and accumulates the
// log1p(|.|)-weighted grade across the 3 channels.
__global__ __launch_bounds__(THREADS) void dct_grade_kernel(
    const float* __restrict__ x,        // [B, NP, C, 32, 32]
    const float* __restrict__ Dm,       // [32, 32] DCT matrix
    const float* __restrict__ filt,     // [4, 32, 32] bandpass filters
    float* __restrict__ out_coeffs,     // [B, NP, C, 32, 32]
    float* __restrict__ out_grades)     // [B, NP]
{
    __shared__ float s_wm[PATCH * PATCH];                     // weight map
    __shared__ float s_T[WAVES_PER_BLOCK][PATCH * PATCH];     // per-wave T scratch

    const int tid = threadIdx.x;

    // Precompute weight_map = sum_k 2^k * filt[k] (block-wide, once).
    for (int i = tid; i < PATCH * PATCH; i += THREADS) {
        float w = 0.0f;
        #pragma unroll
        for (int k = 0; k < KFILT; ++k)
            w += (float)(1 << k) * filt[k * PATCH * PATCH + i];
        s_wm[i] = w;
    }
    __syncthreads();

    const int wave = tid >> 5;     // 0..7
    const int lane = tid & 31;
    const int half = lane >> 4;    // 0 or 1 (which 16-lane half)
    const int idx  = lane & 15;    // 0..15

    // A-fragments of the DCT matrix D: Ad[mt][kb], 16x4 tile at rows mt*16..,
    // K-slice kb*4.. . Per ISA layout: vgpr j, lane l -> D[mt*16 + l%16][kb*4 + 2*(l/16) + j].
    // These same registers also serve as B-fragments of D^T for the second matmul:
    // Bd[kb][nt] element (vgpr j, lane l) = D^T[kb*4+2*half+j][nt*16+idx]
    //                                     = D[nt*16+idx][kb*4+2*half+j] == Ad[nt][kb].
    v2f Ad[2][8];
    #pragma unroll
    for (int mt = 0; mt < 2; ++mt) {
        const float* drow = Dm + (mt * 16 + idx) * PATCH;
        #pragma unroll
        for (int kb = 0; kb < 8; ++kb) {
            Ad[mt][kb].x = drow[kb * 4 + 2 * half + 0];
            Ad[mt][kb].y = drow[kb * 4 + 2 * half + 1];
        }
    }

    const int task = blockIdx.x * WAVES_PER_BLOCK + wave;  // grid is exact: 0..8191
    float grade = 0.0f;
    float* Trow = &s_T[wave][0];

    for (int c = 0; c < CH; ++c) {
        const float* xp = x + ((size_t)task * CH + c) * (PATCH * PATCH);
        float*       yp = out_coeffs + ((size_t)task * CH + c) * (PATCH * PATCH);

        // B-fragments of X: Bx[kb][nt], element (vgpr j, lane l) =
        //   X[kb*4 + 2*(l/16) + j][nt*16 + l%16]
        v2f Bx[8][2];
        #pragma unroll
        for (int kb = 0; kb < 8; ++kb) {
            #pragma unroll
            for (int nt = 0; nt < 2; ++nt) {
                Bx[kb][nt].x = xp[(kb * 4 + 2 * half + 0) * PATCH + nt * 16 + idx];
                Bx[kb][nt].y = xp[(kb * 4 + 2 * half + 1) * PATCH + nt * 16 + idx];
            }
        }

        // ---- T = D @ X  (32 wmma) ----
        v8f acc[2][2] = {v8f{}, v8f{}, v8f{}, v8f{}};
        #pragma unroll
        for (int kb = 0; kb < 8; ++kb) {
            #pragma unroll
            for (int mt = 0; mt < 2; ++mt) {
                #pragma unroll
                for (int nt = 0; nt < 2; ++nt) {
                    acc[mt][nt] = __builtin_amdgcn_wmma_f32_16x16x4_f32(
                        /*neg_a=*/false, Ad[mt][kb],
                        /*neg_b=*/false, Bx[kb][nt],
                        /*c_mod=*/(short)0, acc[mt][nt],
                        /*reuse_a=*/false, /*reuse_b=*/false);
                }
            }
        }

        // Stash T into per-wave LDS (C-layout -> row-major), then reload in
        // A-fragment layout. LDS ops are in-order within a wave.
        #pragma unroll
        for (int mt = 0; mt < 2; ++mt)
            #pragma unroll
            for (int nt = 0; nt < 2; ++nt)
                #pragma unroll
                for (int r = 0; r < 8; ++r)
                    Trow[(mt * 16 + half * 8 + r) * PATCH + nt * 16 + idx] = acc[mt][nt][r];

        v2f At[2][8];
        #pragma unroll
        for (int mt = 0; mt < 2; ++mt) {
            #pragma unroll
            for (int kb = 0; kb < 8; ++kb) {
                At[mt][kb].x = Trow[(mt * 16 + idx) * PATCH + kb * 4 + 2 * half + 0];
                At[mt][kb].y = Trow[(mt * 16 + idx) * PATCH + kb * 4 + 2 * half + 1];
            }
        }

        // ---- Y = T @ D^T  (32 wmma; B-fragments of D^T are Ad[nt][kb]) ----
        v8f y[2][2] = {v8f{}, v8f{}, v8f{}, v8f{}};
        #pragma unroll
        for (int kb = 0; kb < 8; ++kb) {
            #pragma unroll
            for (int mt = 0; mt < 2; ++mt) {
                #pragma unroll
                for (int nt = 0; nt < 2; ++nt) {
                    y[mt][nt] = __builtin_amdgcn_wmma_f32_16x16x4_f32(
                        false, At[mt][kb],
                        false, Ad[nt][kb],
                        (short)0, y[mt][nt],
                        false, false);
                }
            }
        }

        // ---- store coefficients + accumulate grade ----
        #pragma unroll
        for (int mt = 0; mt < 2; ++mt) {
            #pragma unroll
            for (int nt = 0; nt < 2; ++nt) {
                #pragma unroll
                for (int r = 0; r < 8; ++r) {
                    const int m = mt * 16 + half * 8 + r;
                    const int n = nt * 16 + idx;
                    const float v = y[mt][nt][r];
                    yp[m * PATCH + n] = v;
                    grade += log1pf(fabsf(v)) * s_wm[m * PATCH + n];
                }
            }
        }
    }

    // Wave32 reduction of the grade; lane 0 writes.
    #pragma unroll
    for (int off = 16; off > 0; off >>= 1)
        grade += __shfl_down(grade, off, 32);
    if (lane == 0)
        out_grades[task] = grade;
}

extern "C" void kernel_launch(void* const* d_in, const int* in_sizes, int n_in,
                              void* d_out, int out_size, void* d_ws, size_t ws_size,
                              hipStream_t stream) {
    const float* x    = (const float*)d_in[0];   // [32,256,3,32,32]
    const float* Dm   = (const float*)d_in[1];   // [32,32]
    const float* filt = (const float*)d_in[2];   // [4,32,32]

    float* coeffs = (float*)d_out;                                   // 25,165,824 floats
    float* grades = coeffs + (size_t)BATCH * NPAT * CH * PATCH * PATCH;  // 8,192 floats

    dct_grade_kernel<<<TASKS / WAVES_PER_BLOCK, THREADS, 0, stream>>>(
        x, Dm, filt, coeffs, grades);
}